// PositiveDefiniteMatrixGenerator_4329327034392
// MI455X (gfx1250) — compile-verified
//
#include <hip/hip_runtime.h>
#include <hip/hip_bf16.h>
#include <math.h>

typedef __attribute__((ext_vector_type(2))) float v2f;
typedef __attribute__((ext_vector_type(4))) float v4f;
typedef __attribute__((ext_vector_type(8))) float v8f;
typedef __attribute__((ext_vector_type(4))) int   v4i;

#define BATCH    8192
#define IN       512
#define T_PARAMS 2080     // 64*65/2
#define MDIM     64
#define XPAD     516      // 512 + 4 : bank = (4*row + k) & 63, conflict-free
#define LPAD     68       // 64 + 4  : same trick, float4-aligned

#define AS1 __attribute__((address_space(1)))
#define AS3 __attribute__((address_space(3)))

#if __has_builtin(__builtin_amdgcn_global_load_async_to_lds_b128) && \
    __has_builtin(__builtin_amdgcn_global_load_async_to_lds_b32)
#define HAVE_ASYNC_LDS 1
#else
#define HAVE_ASYNC_LDS 0
#endif

static __device__ __forceinline__ void wait_asynccnt0() {
#if __has_builtin(__builtin_amdgcn_s_wait_asynccnt)
  __builtin_amdgcn_s_wait_asynccnt(0);
#else
  asm volatile("s_wait_asynccnt 0" ::: "memory");
#endif
}

#if HAVE_ASYNC_LDS
// memory -> LDS DMA, 16B per lane (ASYNCcnt-tracked)
static __device__ __forceinline__ void async_cp16(const float* gsrc, float* ldst) {
  __builtin_amdgcn_global_load_async_to_lds_b128(
      (AS1 v4i*)(gsrc), (AS3 v4i*)(ldst), 0, 0);
}
// memory -> LDS DMA, 4B per lane
static __device__ __forceinline__ void async_cp4(const float* gsrc, float* ldst) {
  __builtin_amdgcn_global_load_async_to_lds_b32(
      (AS1 int*)(gsrc), (AS3 int*)(ldst), 0, 0);
}
#endif

// -------- Kernel 1: chol = x @ W^T + b, ReLU diag, scatter to L in d_out ----
// Block: 256 threads (8 wave32). Each block owns 16 batch rows; each wave
// iterates over 16-wide N tiles of the 2080 tril parameters.
__global__ __launch_bounds__(256) void pdm_fc_kernel(
    const float* __restrict__ x, const float* __restrict__ W,
    const float* __restrict__ bias, float* __restrict__ out)
{
  __shared__ float xs[16 * XPAD];
  const int tid  = threadIdx.x;
  const int lane = tid & 31;
  const int wave = tid >> 5;
  const int bi0  = blockIdx.x * 16;

  // Stage x tile [16 x 512] -> LDS (padded rows), via async LDS DMA when
  // available (ASYNCcnt path), else through VGPRs.
#if HAVE_ASYNC_LDS
  for (int q = tid; q < (16 * IN) / 4; q += 256) {
    int row = q >> 7;          // 128 float4 per row
    int c4  = (q & 127) * 4;
    async_cp16(x + (size_t)(bi0 + row) * IN + c4, xs + row * XPAD + c4);
  }
  wait_asynccnt0();
#else
  for (int q = tid; q < (16 * IN) / 4; q += 256) {
    int row = q >> 7;
    int c4  = (q & 127) * 4;
    v4f g = *(const v4f*)(x + (size_t)(bi0 + row) * IN + c4);
    *(v4f*)(xs + row * XPAD + c4) = g;
  }
#endif
  __syncthreads();

  const int mrow  = lane & 15;   // M row for A-frag, N col for B/C-frag
  const int khalf = lane >> 4;   // 0 -> K{0,1}, 1 -> K{2,3}

  for (int tile = wave; tile < T_PARAMS / 16; tile += 8) {
    const int n0 = tile * 16;
    const int t  = n0 + mrow;                       // this lane's param index
    const float* wrow = W + (size_t)t * IN;         // B-frag source (row of W)
    if (t + 128 < T_PARAMS)                         // warm L2/L0 for next tile
      __builtin_prefetch(W + (size_t)(t + 128) * IN, 0, 1);

    v8f acc = {};
#pragma unroll 8
    for (int k0 = 0; k0 < IN; k0 += 4) {
      v2f a  = *(const v2f*)(xs + mrow * XPAD + k0 + 2 * khalf);
      v2f bf = *(const v2f*)(wrow + k0 + 2 * khalf);
      acc = __builtin_amdgcn_wmma_f32_16x16x4_f32(false, a, false, bf,
                                                  (short)0, acc, false, false);
    }

    // row-major lower-tri index -> (i, j)
    float ft = sqrtf(8.0f * (float)t + 1.0f);
    int i = (int)((ft - 1.0f) * 0.5f);
    while ((i + 1) * (i + 2) / 2 <= t) ++i;
    while (i * (i + 1) / 2 > t) --i;
    const int j = t - i * (i + 1) / 2;
    const float bv = bias[t];

#pragma unroll
    for (int r = 0; r < 8; ++r) {
      const int batch = bi0 + r + 8 * khalf;        // C layout: VGPR r = M r/r+8
      float v = acc[r] + bv;
      if (i == j) v = fmaxf(v, 0.0f);
      out[(size_t)batch * (MDIM * MDIM) + i * MDIM + j] = v;
    }
  }
}

// -------- Kernel 2: per-batch L @ L^T, in place on d_out ---------------------
// 8 batches per block staged in LDS (upper triangle masked to 0 at load).
__global__ __launch_bounds__(256) void pdm_llt_kernel(float* __restrict__ out)
{
  __shared__ float Ls[8 * MDIM * LPAD];
  const int tid  = threadIdx.x;
  const int lane = tid & 31;
  const int wave = tid >> 5;
  const int b0   = blockIdx.x * 8;

  // Load 8 x (64x64) L tiles, zeroing cols > row (K1 only wrote the tril part)
  for (int q = tid; q < 8 * 1024; q += 256) {       // 1024 float4 per batch
    const int bl  = q >> 10;
    const int rem = q & 1023;
    const int row = rem >> 4;                       // 16 float4 per row
    const int c4  = (rem & 15) * 4;
    float*       dst = Ls + bl * (MDIM * LPAD) + row * LPAD + c4;
    const float* src = out + ((size_t)(b0 + bl) << 12) + row * MDIM + c4;
#if HAVE_ASYNC_LDS
    if (c4 + 3 <= row) {
      // fully inside lower triangle: straight memory->LDS DMA
      async_cp16(src, dst);
    } else if (c4 > row) {
      // fully upper: just zero (async never touches these words -> no race)
      *(v4f*)dst = (v4f){0.0f, 0.0f, 0.0f, 0.0f};
    } else {
      // straddles the diagonal: per-word DMA / zero on disjoint words
#pragma unroll
      for (int c = 0; c < 4; ++c) {
        if (c4 + c <= row)
          async_cp4(src + c, dst + c);
        else
          dst[c] = 0.0f;
      }
    }
#else
    v4f g = *(const v4f*)src;
#pragma unroll
    for (int c = 0; c < 4; ++c)
      if (c4 + c > row) g[c] = 0.0f;
    *(v4f*)dst = g;
#endif
  }
#if HAVE_ASYNC_LDS
  wait_asynccnt0();   // each wave drains its own ASYNCcnt, then barrier
#endif
  __syncthreads();

  const int mrow  = lane & 15;
  const int khalf = lane >> 4;

  // 8 batches * 16 (16x16) output tiles = 128 jobs over 8 waves
  for (int job = wave; job < 128; job += 8) {
    const int bl   = job >> 4;
    const int tile = job & 15;
    const int i0   = (tile >> 2) * 16;
    const int j0   = (tile & 3) * 16;
    const float* Lb = Ls + bl * (MDIM * LPAD);

    v8f acc = {};
#pragma unroll
    for (int k0 = 0; k0 < MDIM; k0 += 4) {
      // (L L^T)[i][j] = sum_k L[i][k] * L[j][k] : both frags are rows of L
      v2f a  = *(const v2f*)(Lb + (i0 + mrow) * LPAD + k0 + 2 * khalf);
      v2f bf = *(const v2f*)(Lb + (j0 + mrow) * LPAD + k0 + 2 * khalf);
      acc = __builtin_amdgcn_wmma_f32_16x16x4_f32(false, a, false, bf,
                                                  (short)0, acc, false, false);
    }

    float* ob = out + ((size_t)(b0 + bl) << 12);
#pragma unroll
    for (int r = 0; r < 8; ++r)
      ob[(i0 + r + 8 * khalf) * MDIM + j0 + mrow] = acc[r];
  }
}

extern "C" void kernel_launch(void* const* d_in, const int* in_sizes, int n_in,
                              void* d_out, int out_size, void* d_ws, size_t ws_size,
                              hipStream_t stream) {
  (void)in_sizes; (void)n_in; (void)out_size; (void)d_ws; (void)ws_size;
  const float* x    = (const float*)d_in[0];
  const float* W    = (const float*)d_in[1];
  const float* bias = (const float*)d_in[2];
  float* out = (float*)d_out;

  pdm_fc_kernel<<<BATCH / 16, 256, 0, stream>>>(x, W, bias, out);
  pdm_llt_kernel<<<BATCH / 8, 256, 0, stream>>>(out);
}